// CSAM_Module_69045894250745
// MI455X (gfx1250) — compile-verified
//
#include <hip/hip_runtime.h>

typedef __attribute__((ext_vector_type(2))) float v2f;
typedef __attribute__((ext_vector_type(8))) float v8f;
typedef __attribute__((ext_vector_type(4))) int   v4i;

constexpr int B_ = 8, N_ = 7, C_ = 64, H_ = 128, W_ = 128;
constexpr int D_ = C_ * H_ * W_;        // 1,048,576 floats per (b,n)
constexpr int CHUNK = 1024;             // d-range per block (floats)
constexpr int NCHUNK = D_ / CHUNK;      // 1024
constexpr int LDS_STRIDE = CHUNK + 4;   // stride % 64 == 4 -> conflict-free frags

#if defined(__has_builtin)
# if __has_builtin(__builtin_amdgcn_global_load_async_to_lds_b128)
#  define HAVE_ASYNC_LDS 1
# endif
# if __has_builtin(__builtin_amdgcn_s_wait_asynccnt)
#  define HAVE_WAIT_ASYNC 1
# endif
#endif

// ---------------------------------------------------------------------------
// Phase 1: partial Gram energy[b][n][m] += sum_d v[b,n,d]*v[b,m,d]
// via V_WMMA_F32_16X16X4_F32. 8 LDS rows: rows 0-6 = data, row 7 = zeros;
// padded lanes (row>=7) clamp to the zero row so all loads are unconditional.
// ---------------------------------------------------------------------------
__global__ __launch_bounds__(256) void gram_wmma_kernel(
    const float* __restrict__ x2, float* __restrict__ energy)
{
    __shared__ float tile[8 * LDS_STRIDE];    // ~32 KB
    __shared__ float red[8 * 49];             // per-wave partial results

    const int tid = threadIdx.x;
    const int b  = blockIdx.y;
    const int d0 = blockIdx.x * CHUNK;

    // zero the padding row (row 7)
    for (int i = tid; i < LDS_STRIDE; i += 256)
        tile[7 * LDS_STRIDE + i] = 0.f;

    // Stage the 7 x CHUNK tile into LDS, 16B per lane, coalesced.
#ifdef HAVE_ASYNC_LDS
    for (int i = tid; i < N_ * (CHUNK / 4); i += 256) {
        const int r  = i >> 8;               // / (CHUNK/4 == 256)
        const int c4 = i & 255;
        const float* g = x2 + ((size_t)b * N_ + r) * D_ + d0 + (c4 << 2);
        float* l = &tile[r * LDS_STRIDE + (c4 << 2)];
        __builtin_amdgcn_global_load_async_to_lds_b128((v4i*)g, (v4i*)l, 0, 0);
    }
# ifdef HAVE_WAIT_ASYNC
    __builtin_amdgcn_s_wait_asynccnt(0);
# else
    asm volatile("s_wait_asynccnt 0x0" ::: "memory");
# endif
#else
    const float4* src = reinterpret_cast<const float4*>(x2);
    for (int i = tid; i < N_ * (CHUNK / 4); i += 256) {
        const int r  = i >> 8;
        const int c4 = i & 255;
        float4 v = src[((size_t)b * N_ + r) * (D_ / 4) + (d0 >> 2) + c4];
        *reinterpret_cast<float4*>(&tile[r * LDS_STRIDE + (c4 << 2)]) = v;
    }
#endif
    __syncthreads();

    // Fragment mapping for 32-bit 16x4 A (B identical for a Gram):
    // lanes 0-15: row = lane, K = {0,1}; lanes 16-31: row = lane-16, K = {2,3}
    const int wave  = tid >> 5;
    const int lane  = tid & 31;
    const int row   = lane & 15;
    const int khalf = lane >> 4;
    const int rclmp = (row < N_) ? row : 7;       // padded lanes -> zero row
    const float* tbase = &tile[rclmp * LDS_STRIDE + 2 * khalf];

    v8f acc0 = {0.f,0.f,0.f,0.f,0.f,0.f,0.f,0.f};
    v8f acc1 = {0.f,0.f,0.f,0.f,0.f,0.f,0.f,0.f};
    const int dbeg = wave * (CHUNK / 8);          // 128 d-values per wave
#pragma unroll 4
    for (int d = dbeg; d < dbeg + CHUNK / 8; d += 8) {
        v2f a0 = *reinterpret_cast<const v2f*>(tbase + d);
        v2f a1 = *reinterpret_cast<const v2f*>(tbase + d + 4);
        acc0 = __builtin_amdgcn_wmma_f32_16x16x4_f32(
            false, a0, false, a0, (short)0, acc0, false, false);
        acc1 = __builtin_amdgcn_wmma_f32_16x16x4_f32(
            false, a1, false, a1, (short)0, acc1, false, false);
    }

    // C layout: VGPR j, lanes 0-15 -> (M=j, N=lane). Valid region M<7, N<7.
    if (khalf == 0 && row < N_) {
#pragma unroll
        for (int j = 0; j < N_; ++j)
            red[wave * 49 + j * 7 + row] = acc0[j] + acc1[j];
    }
    __syncthreads();

    if (tid < 49) {
        float s = 0.f;
#pragma unroll
        for (int w = 0; w < 8; ++w) s += red[w * 49 + tid];
        atomicAdd(&energy[b * 49 + tid], s);
    }
}

// ---------------------------------------------------------------------------
// Phase 2: attg[b][n][m] = gamma * softmax_m( rowmax(energy) - energy )
// ---------------------------------------------------------------------------
__global__ void softmax_gamma_kernel(const float* __restrict__ energy,
                                     const float* __restrict__ gamma,
                                     float* __restrict__ attg)
{
    const int t = threadIdx.x;
    if (t >= B_ * N_) return;
    const int b = t / N_, n = t % N_;
    const float* e = energy + b * 49 + n * 7;

    float mx = e[0];
#pragma unroll
    for (int m = 1; m < N_; ++m) mx = fmaxf(mx, e[m]);
    float en[N_]; float mx2 = -3.4e38f;
#pragma unroll
    for (int m = 0; m < N_; ++m) { en[m] = mx - e[m]; mx2 = fmaxf(mx2, en[m]); }
    float s = 0.f;
#pragma unroll
    for (int m = 0; m < N_; ++m) { float ex = __expf(en[m] - mx2); en[m] = ex; s += ex; }
    const float g = gamma[0] / s;
#pragma unroll
    for (int m = 0; m < N_; ++m) attg[b * 49 + n * 7 + m] = en[m] * g;
}

// ---------------------------------------------------------------------------
// Phase 3 (bandwidth-dominant): out[b,n,d] = x1 * (sum_m attg[n][m]*v[b,m,d]) + x1
// ---------------------------------------------------------------------------
__global__ __launch_bounds__(256) void apply_kernel(
    const float* __restrict__ x1, const float* __restrict__ x2,
    const float* __restrict__ attg, float* __restrict__ out)
{
    __shared__ float a[49];
    const int b = blockIdx.y;
    if (threadIdx.x < 49) a[threadIdx.x] = attg[b * 49 + threadIdx.x];
    __syncthreads();

    const size_t d4   = (size_t)blockIdx.x * blockDim.x + threadIdx.x; // < D_/4
    const size_t base = (size_t)b * N_ * (D_ / 4) + d4;
    const float4* X2 = reinterpret_cast<const float4*>(x2) + base;
    const float4* X1 = reinterpret_cast<const float4*>(x1) + base;
    float4*       O  = reinterpret_cast<float4*>(out) + base;

    float4 vm[N_];
#pragma unroll
    for (int m = 0; m < N_; ++m) vm[m] = X2[(size_t)m * (D_ / 4)];

#pragma unroll
    for (int n = 0; n < N_; ++n) {
        float4 s; s.x = s.y = s.z = s.w = 0.f;
#pragma unroll
        for (int m = 0; m < N_; ++m) {
            const float w = a[n * 7 + m];
            s.x = fmaf(w, vm[m].x, s.x);
            s.y = fmaf(w, vm[m].y, s.y);
            s.z = fmaf(w, vm[m].z, s.z);
            s.w = fmaf(w, vm[m].w, s.w);
        }
        float4 x = X1[(size_t)n * (D_ / 4)];
        float4 r;
        r.x = fmaf(x.x, s.x, x.x);
        r.y = fmaf(x.y, s.y, x.y);
        r.z = fmaf(x.z, s.z, x.z);
        r.w = fmaf(x.w, s.w, x.w);
        O[(size_t)n * (D_ / 4)] = r;
    }
}

// ---------------------------------------------------------------------------
extern "C" void kernel_launch(void* const* d_in, const int* in_sizes, int n_in,
                              void* d_out, int out_size, void* d_ws, size_t ws_size,
                              hipStream_t stream)
{
    const float* x1    = (const float*)d_in[0];
    const float* x2    = (const float*)d_in[1];
    const float* gamma = (const float*)d_in[2];
    float* out = (float*)d_out;

    float* energy = (float*)d_ws;          // B*49 floats
    float* attg   = energy + B_ * 49;      // B*49 floats

    (void)hipMemsetAsync(d_ws, 0, (size_t)B_ * 49 * sizeof(float), stream);

    dim3 g1(NCHUNK, B_);
    gram_wmma_kernel<<<g1, 256, 0, stream>>>(x2, energy);

    softmax_gamma_kernel<<<1, 64, 0, stream>>>(energy, gamma, attg);

    dim3 g3((D_ / 4) / 256, B_);
    apply_kernel<<<g3, 256, 0, stream>>>(x1, x2, attg, out);
}